// AGNNConv_43396349558834
// MI455X (gfx1250) — compile-verified
//
#include <hip/hip_runtime.h>
#include <hip/hip_bf16.h>

// ---------------------------------------------------------------------------
// AGNNConv for MI455X (gfx1250, wave32).
//   Phase 1: X_prime = X @ W  via V_WMMA_F32_16X16X4_F32 (fp32 tensor path)
//   Phase 2: per-edge dot + attention + segment_sum over sorted edge_src,
//            register-accumulated per src-run, flushed with fp32 atomics.
//            src-feature load hoisted to run boundaries (halves gather BW).
// ---------------------------------------------------------------------------

typedef __attribute__((ext_vector_type(2))) float v2f;
typedef __attribute__((ext_vector_type(8))) float v8f;

#define N_NODES 100000
#define IN_DIM  128
#define OUT_DIM 64
#define EDGES_PER_WAVE 256

// ---------------------------------------------------------------------------
// Zero-init output (harness poisons d_out with 0xAA).
// ---------------------------------------------------------------------------
__global__ void zero_f32(float* __restrict__ p, int n) {
    int i = blockIdx.x * blockDim.x + threadIdx.x;
    if (i < n) p[i] = 0.0f;
}

// ---------------------------------------------------------------------------
// GEMM: XP[100000 x 64] = X[100000 x 128] @ W[128 x 64], fp32 WMMA.
// One wave computes one 16x16 output tile; blockDim=128 (4 waves), each wave
// takes a different N-tile so the block shares the same 16 X rows in cache.
//
// VGPR layouts (ISA 7.12.2, 32-bit matrices, wave32):
//   A 16x4:  lane L (0-15): M=L, K={k0,k0+1};  lane L+16: M=L, K={k0+2,k0+3}
//            -> per step one aligned float2 load per lane.
//   B 4x16:  VGPR0: rows K=k0 (lanes 0-15) / K=k0+2 (lanes 16-31), N=lane&15
//            VGPR1: rows K=k0+1 / K=k0+3.
//   C 16x16: VGPR v: M = v + (lane>>4)*8, N = lane&15.
// ---------------------------------------------------------------------------
__global__ void __launch_bounds__(128)
gemm_xw_wmma(const float* __restrict__ X, const float* __restrict__ W,
             float* __restrict__ XP) {
    const int lane  = threadIdx.x & 31;
    const int tileN = threadIdx.x >> 5;          // 0..3  (OUT_DIM/16)
    const int tileM = blockIdx.x;                // 0..6249 (N_NODES/16)
    const int row   = lane & 15;
    const int khalf = lane >> 4;                 // 0 or 1

    const float* __restrict__ Xrow = X + (size_t)(tileM * 16 + row) * IN_DIM;
    const float* __restrict__ Wcol = W + tileN * 16 + row;   // N = row for B

    v8f c = {};
#pragma unroll
    for (int k0 = 0; k0 < IN_DIM; k0 += 4) {
        // A: contiguous K pair for this lane's half -> single b64 load
        v2f a = *(const v2f*)(Xrow + k0 + khalf * 2);
        // B: rows k0+khalf*2 and k0+khalf*2+1, column (lane&15) of W tile
        v2f b;
        b.x = Wcol[(size_t)(k0 + khalf * 2 + 0) * OUT_DIM];
        b.y = Wcol[(size_t)(k0 + khalf * 2 + 1) * OUT_DIM];
        // (neg_a, A, neg_b, B, c_mod, C, reuse_a, reuse_b)
        c = __builtin_amdgcn_wmma_f32_16x16x4_f32(
                false, a, false, b, (short)0, c, false, false);
    }

    float* __restrict__ outp = XP + (size_t)(tileM * 16) * OUT_DIM
                                  + tileN * 16 + (lane & 15);
#pragma unroll
    for (int v = 0; v < 8; ++v) {
        int m = v + khalf * 8;
        outp[(size_t)m * OUT_DIM] = c[v];
    }
}

// ---------------------------------------------------------------------------
// Edge phase. One wave owns EDGES_PER_WAVE contiguous (src-sorted) edges.
// Lane L holds output columns {2L, 2L+1} (64 cols / 32 lanes).
// Running accumulator per src-run in registers; src feature loaded once per
// run; flush with fp32 atomics only when src changes (runs average ~16 edges
// -> ~64x fewer atomics than naive, and ~2x less gather traffic than
// reloading src per edge).
// ---------------------------------------------------------------------------
__global__ void __launch_bounds__(256)
agnn_edge_kernel(const float* __restrict__ XP,
                 const int* __restrict__ esrc,
                 const int* __restrict__ edst,
                 const float* __restrict__ attw,
                 float* __restrict__ out, int nEdges) {
    const int lane = threadIdx.x & 31;
    const int wid  = blockIdx.x * (blockDim.x >> 5) + (threadIdx.x >> 5);

    int e0 = wid * EDGES_PER_WAVE;
    if (e0 >= nEdges) return;
    int e1 = e0 + EDGES_PER_WAVE;
    if (e1 > nEdges) e1 = nEdges;

    const float aw = attw[0];

    int   cur  = -1;
    float acc0 = 0.0f, acc1 = 0.0f;
    v2f   sf   = {0.0f, 0.0f};                   // features of current src

    for (int e = e0; e < e1; ++e) {
        // e is wave-uniform -> compiler scalarizes these to s_load
        const int s = esrc[e];
        const int d = edst[e];

        if (s != cur) {
            if (cur >= 0) {
                float* dst = out + (size_t)cur * OUT_DIM + 2 * lane;
                unsafeAtomicAdd(dst + 0, acc0);
                unsafeAtomicAdd(dst + 1, acc1);
            }
            cur  = s;
            acc0 = 0.0f;
            acc1 = 0.0f;
            // src features loaded once per run (edge_src is sorted)
            sf = *(const v2f*)(XP + (size_t)s * OUT_DIM + 2 * lane);
        }

        const v2f df = *(const v2f*)(XP + (size_t)d * OUT_DIM + 2 * lane);

        float p = sf.x * df.x + sf.y * df.y;
#pragma unroll
        for (int m = 16; m > 0; m >>= 1)
            p += __shfl_xor(p, m, 32);          // wave32 butterfly reduce

        const float att = p * aw;
        acc0 += att * df.x;
        acc1 += att * df.y;
    }

    if (cur >= 0) {
        float* dst = out + (size_t)cur * OUT_DIM + 2 * lane;
        unsafeAtomicAdd(dst + 0, acc0);
        unsafeAtomicAdd(dst + 1, acc1);
    }
}

// ---------------------------------------------------------------------------
// Launch: inputs are {X, weights, attention_w, edge_src, edge_dst}.
// d_ws holds X_prime (100000*64 fp32 = 25.6 MB).
// ---------------------------------------------------------------------------
extern "C" void kernel_launch(void* const* d_in, const int* in_sizes, int n_in,
                              void* d_out, int out_size, void* d_ws, size_t ws_size,
                              hipStream_t stream) {
    const float* X    = (const float*)d_in[0];
    const float* W    = (const float*)d_in[1];
    const float* attw = (const float*)d_in[2];
    const int*   esrc = (const int*)d_in[3];
    const int*   edst = (const int*)d_in[4];
    float*       out  = (float*)d_out;
    float*       XP   = (float*)d_ws;

    const int nEdges = in_sizes[3];

    // 1) zero the output accumulator
    {
        int n = out_size;
        zero_f32<<<(n + 255) / 256, 256, 0, stream>>>(out, n);
    }

    // 2) X_prime = X @ W  (6250 M-tiles x 4 N-tiles, one tile per wave)
    gemm_xw_wmma<<<N_NODES / 16, 128, 0, stream>>>(X, W, XP);

    // 3) fused edge-dot + attention + segment-sum
    {
        int waves  = (nEdges + EDGES_PER_WAVE - 1) / EDGES_PER_WAVE;
        int blocks = (waves + 7) / 8;            // 8 waves / 256-thread block
        agnn_edge_kernel<<<blocks, 256, 0, stream>>>(XP, esrc, edst, attw,
                                                     out, nEdges);
    }
}